// PredictCenter_84293028151785
// MI455X (gfx1250) — compile-verified
//
#include <hip/hip_runtime.h>
#include <hip/hip_bf16.h>
#include <math.h>

// Problem constants (match reference setup_inputs)
#define BB    16
#define CC    80
#define HH    160
#define WW    160
#define HW    25600          // HH*WW
#define KTOP  70
#define THR_CONF 0.5f
#define THR_NMS  0.3f

// ---------------------------------------------------------------------------
// Async global->LDS helpers (CDNA5 ASYNCcnt path)
// ---------------------------------------------------------------------------
__device__ __forceinline__ void async_g2l_b128(unsigned lds_byte,
                                               unsigned long long gaddr)
{
    asm volatile("global_load_async_to_lds_b128 %0, %1, off"
                 :: "v"(lds_byte), "v"(gaddr) : "memory");
}

// Copy `bytes` (multiple of 16, 16B-aligned both sides) with 256 threads.
__device__ __forceinline__ void async_copy_g2l(void* ldst, const void* gsrc,
                                               unsigned bytes)
{
    const unsigned           l = (unsigned)(size_t)ldst;
    const unsigned long long g = (unsigned long long)(size_t)gsrc;
    for (unsigned off = threadIdx.x * 16u; off < bytes; off += 4096u)
        async_g2l_b128(l + off, g + (unsigned long long)off);
}

// ---------------------------------------------------------------------------
// 3x3 local-max test via sliding column-max:
//  local-max(v at (row,col))  <=>  v >= max(colMax(col-1), colMax(col), colMax(col+1))
// where colMax(c) = max over valid rows {row-1,row,row+1} at column c.
// ---------------------------------------------------------------------------
__device__ __forceinline__ unsigned mask_word(const float* pl, int w)
{
    const int  row     = w / 5;
    const int  colBase = (w % 5) * 32;        // words never cross a row
    const int  rbase   = row * WW;
    const bool up = (row > 0), dn = (row < HH - 1);

    auto colmax = [&](int c, float& center) -> float {
        float v = pl[rbase + c];
        center = v;
        float m = v;
        if (up) m = fmaxf(m, pl[rbase - WW + c]);
        if (dn) m = fmaxf(m, pl[rbase + WW + c]);
        return m;
    };

    float dummy, vcur;
    float cm_m1 = (colBase > 0) ? colmax(colBase - 1, dummy) : -INFINITY;
    float cm_0  = colmax(colBase, vcur);
    unsigned bits = 0u;
    #pragma unroll 4
    for (int j = 0; j < 32; ++j) {
        const int cnext = colBase + j + 1;
        float vnext = 0.0f;
        const float cm_p1 = (cnext < WW) ? colmax(cnext, vnext) : -INFINITY;
        if (vcur >= cm_m1 && vcur >= cm_0 && vcur >= cm_p1) bits |= (1u << j);
        cm_m1 = cm_0; cm_0 = cm_p1; vcur = vnext;
    }
    return bits;
}

// ---------------------------------------------------------------------------
// Stage 1: one block per (b,c) plane.
//  - async-copy 100KB plane HBM -> LDS, pipelined against mask computation
//  - 3x3 local-max bitmask
//  - extract plane top-70 via incremental block argmax
// ---------------------------------------------------------------------------
__global__ __launch_bounds__(256) void
pc_stage1_pool_topk(const float* __restrict__ heat,
                    float* __restrict__ candS,
                    int*   __restrict__ candI)
{
    extern __shared__ char smem[];
    float*    pl   = (float*)    (smem);               // 102400 B : raw plane
    unsigned* msk  = (unsigned*) (smem + 102400);      //   3200 B : local-max bits
    float*    lmax = (float*)    (smem + 105600);      //   1024 B
    int*      lidx = (int*)      (smem + 106624);      //   1024 B
    float*    redv = (float*)    (smem + 107648);      //   1024 B
    int*      redi = (int*)      (smem + 108672);      //   1024 B
    float*    outS = (float*)    (smem + 109696);      //    280 B
    int*      outI = (int*)      (smem + 109976);      //    280 B  (total 110256)

    const int t     = threadIdx.x;                     // 0..255
    const int plane = blockIdx.x;                      // b*CC + c
    const float* gp = heat + (size_t)plane * HW;

    // ---- issue all 25 async chunks (25 x 256 x 16B = 102400 B) ------------
    {
        const unsigned           lbase = (unsigned)(size_t)pl;
        const unsigned long long gbase = (unsigned long long)(size_t)gp;
        #pragma unroll
        for (int i = 0; i < 25; ++i) {
            const unsigned off = (unsigned)(t * 16 + i * 4096);
            async_g2l_b128(lbase + off, gbase + (unsigned long long)off);
        }
    }

    // ---- phase 1: first 13 chunks (elements < 13312 => rows 0..82 full) ---
    // async loads complete in issue order per wave; cnt<=12 => chunks 0..12 done
    asm volatile("s_wait_asynccnt 12" ::: "memory");
    __syncthreads();
    for (int w = t; w < 410; w += 256)     // mask rows 0..81 (need row r+1<=82)
        msk[w] = mask_word(pl, w);

    // ---- phase 2: remaining rows ------------------------------------------
    asm volatile("s_wait_asynccnt 0" ::: "memory");
    __syncthreads();
    for (int w = 410 + t; w < 800; w += 256)
        msk[w] = mask_word(pl, w);
    __syncthreads();

    // ---- per-thread local argmax over its 100 strided elements -------------
    {
        float bv = -INFINITY; int bi = t;
        for (int k = 0; k < 100; ++k) {
            const int e = t + (k << 8);
            const unsigned bit = (msk[e >> 5] >> (e & 31)) & 1u;
            const float v = bit ? pl[e] : 0.0f;     // phm = v if local max else 0
            if (v > bv) { bv = v; bi = e; }
        }
        lmax[t] = bv; lidx[t] = bi;
    }
    __syncthreads();

    // ---- extract top-70: only the winner rescans its slice each round ------
    for (int i = 0; i < KTOP; ++i) {
        redv[t] = lmax[t]; redi[t] = t;
        __syncthreads();
        for (int s = 128; s > 0; s >>= 1) {
            if (t < s && redv[t + s] > redv[t]) { redv[t] = redv[t + s]; redi[t] = redi[t + s]; }
            __syncthreads();
        }
        const int   wn = redi[0];
        const float bv = redv[0];
        if (t == wn) {
            const int e = lidx[t];
            outS[i] = bv; outI[i] = e;
            pl[e] = -INFINITY;                      // consume (bit stays set)
            float nv = -INFINITY; int ni = t;
            for (int k = 0; k < 100; ++k) {
                const int e2 = t + (k << 8);
                const unsigned bit = (msk[e2 >> 5] >> (e2 & 31)) & 1u;
                const float v = bit ? pl[e2] : 0.0f;
                if (v > nv) { nv = v; ni = e2; }
            }
            lmax[t] = nv; lidx[t] = ni;
        }
        __syncthreads();
    }

    if (t < KTOP) {
        const int c = plane % CC;
        candS[(size_t)plane * KTOP + t] = outS[t];
        candI[(size_t)plane * KTOP + t] = c * HW + outI[t];   // id within image
    }
}

// ---------------------------------------------------------------------------
// Stage 2: one block per batch image. Merge 80x70 candidates -> exact top-70,
// decode boxes, greedy per-class NMS, write all 5 outputs.
// ---------------------------------------------------------------------------
#define NCAND (CC * KTOP)   // 5600  (5600*4 B = 22400 B, multiple of 16)

__global__ __launch_bounds__(256) void
pc_stage2_merge_nms(const float* __restrict__ candS,
                    const int*   __restrict__ candI,
                    const float* __restrict__ pwh,
                    const float* __restrict__ pxy,
                    float* __restrict__ out)
{
    __shared__ __align__(16) float cs[NCAND];
    __shared__ __align__(16) int   ci[NCAND];
    __shared__ float lmax[256];  __shared__ int lidx[256];
    __shared__ float redv[256];  __shared__ int redi[256];
    __shared__ float tS[KTOP];   __shared__ int tI[KTOP];
    __shared__ float bx[KTOP][4];
    __shared__ float area[KTOP];
    __shared__ int   lab[KTOP];
    __shared__ int   keepf[KTOP];

    const int t = threadIdx.x;
    const int b = blockIdx.x;

    // async-stage both candidate arrays (2 x 22400 B)
    async_copy_g2l(cs, candS + (size_t)b * NCAND, NCAND * 4u);
    async_copy_g2l(ci, candI + (size_t)b * NCAND, NCAND * 4u);
    asm volatile("s_wait_asynccnt 0" ::: "memory");
    __syncthreads();

    {
        float bv = -INFINITY; int bi = 0;
        for (int e = t; e < NCAND; e += 256)
            if (cs[e] > bv) { bv = cs[e]; bi = e; }
        lmax[t] = bv; lidx[t] = bi;
    }
    __syncthreads();

    for (int i = 0; i < KTOP; ++i) {
        redv[t] = lmax[t]; redi[t] = t;
        __syncthreads();
        for (int s = 128; s > 0; s >>= 1) {
            if (t < s && redv[t + s] > redv[t]) { redv[t] = redv[t + s]; redi[t] = redi[t + s]; }
            __syncthreads();
        }
        const int wn = redi[0]; const float v = redv[0];
        if (t == wn) {
            const int e = lidx[t];
            tS[i] = v; tI[i] = ci[e];
            cs[e] = -INFINITY;
            float nv = -INFINITY; int ni = 0;
            for (int e2 = t; e2 < NCAND; e2 += 256)
                if (cs[e2] > nv) { nv = cs[e2]; ni = e2; }
            lmax[t] = nv; lidx[t] = ni;
        }
        __syncthreads();
    }

    // ---- decode boxes (threads 0..69) --------------------------------------
    if (t < KTOP) {
        const int id  = tI[t];
        const int cls = id / HW;
        const int off = id % HW;
        const int row = off / WW;
        const int col = off - row * WW;
        const size_t base = (size_t)b * 2 * HW;
        const float ox = pxy[base + 0 * HW + off];   // channel 0 -> x
        const float oy = pxy[base + 1 * HW + off];   // channel 1 -> y
        const float w0 = pwh[base + 0 * HW + off];
        const float h0 = pwh[base + 1 * HW + off];
        const float x  = ((float)col + ox) * (1.0f / (float)HH);  // grid=[H,W]
        const float y  = ((float)row + oy) * (1.0f / (float)WW);
        const float hw_ = w0 * 0.5f, hh_ = h0 * 0.5f;
        bx[t][0] = x - hw_; bx[t][1] = y - hh_;
        bx[t][2] = x + hw_; bx[t][3] = y + hh_;
        area[t]  = (bx[t][2] - bx[t][0]) * (bx[t][3] - bx[t][1]);
        lab[t]   = cls + 1;
        keepf[t] = (tS[t] > THR_CONF) ? 1 : 0;
    }
    __syncthreads();

    // ---- greedy NMS: keep &= ~( iou>thr & same label & j>i & keep[i] ) -----
    for (int i = 0; i < KTOP; ++i) {
        const int   ki = keepf[i];
        const float l = bx[i][0], tp = bx[i][1], r = bx[i][2], bt = bx[i][3];
        const float ai = area[i];
        const int   li = lab[i];
        if (t < KTOP && t > i && ki && lab[t] == li) {
            float iw = fminf(r,  bx[t][2]) - fmaxf(l,  bx[t][0]); iw = iw > 0.f ? iw : 0.f;
            float ih = fminf(bt, bx[t][3]) - fmaxf(tp, bx[t][1]); ih = ih > 0.f ? ih : 0.f;
            const float inter = iw * ih;
            const float iou = inter / (ai + area[t] - inter + 1e-9f);
            if (iou > THR_NMS) keepf[t] = 0;
        }
        __syncthreads();
    }

    // ---- outputs: [ids 1120][ltrb 4480][labels 1120][scores 1120][keep 1120]
    if (t < KTOP) {
        const int g = b * KTOP + t;
        out[g] = (float)b;
        out[1120 + g * 4 + 0] = bx[t][0];
        out[1120 + g * 4 + 1] = bx[t][1];
        out[1120 + g * 4 + 2] = bx[t][2];
        out[1120 + g * 4 + 3] = bx[t][3];
        out[5600 + g] = (float)lab[t];
        out[6720 + g] = tS[t];
        out[7840 + g] = (float)keepf[t];
    }
}

// ---------------------------------------------------------------------------
extern "C" void kernel_launch(void* const* d_in, const int* in_sizes, int n_in,
                              void* d_out, int out_size, void* d_ws, size_t ws_size,
                              hipStream_t stream)
{
    const float* heat = (const float*)d_in[0];   // [16,80,160,160]
    const float* pwh  = (const float*)d_in[1];   // [16,2,160,160]
    const float* pxy  = (const float*)d_in[2];   // [16,2,160,160]
    // d_in[3] (pkeypoint_offset) is unused by the reference output
    float* out = (float*)d_out;

    // workspace: candS (1280*70 f32) then candI (1280*70 i32)  ~717 KB
    float* candS = (float*)d_ws;
    int*   candI = (int*)((char*)d_ws + (size_t)BB * CC * KTOP * sizeof(float));

    const size_t sh1 = 110256;  // plane + mask + reduction scratch (fits 320KB WGP LDS)
    pc_stage1_pool_topk<<<BB * CC, 256, sh1, stream>>>(heat, candS, candI);
    pc_stage2_merge_nms<<<BB, 256, 0, stream>>>(candS, candI, pwh, pxy, out);
}